// Render_78907139162146
// MI455X (gfx1250) — compile-verified
//
#include <hip/hip_runtime.h>

typedef __attribute__((ext_vector_type(2))) float v2f;
typedef __attribute__((ext_vector_type(8))) float v8f;

#define NTRI 256

__global__ __launch_bounds__(32) void raster16x16_wmma(
    const float* __restrict__ tris,   // [256][3][3]
    const float* __restrict__ uvs,    // [256][3][2]
    const float* __restrict__ tex,    // [3][1024][1024]
    float* __restrict__ out)          // [4][256][256]
{
    // Per-triangle record: 5 x float4, read with ds_load_b128.
    //  q0 = (w1cx, w1cy, w1c0, valid)
    //  q1 = (w2cx, w2cy, w2c0, w3cx)
    //  q2 = (w3cy, w3c0, zcx,  zcy )
    //  q3 = (zc0,  ucx,  ucy,  uc0 )
    //  q4 = (vcx,  vcy,  vc0,  0   )
    __shared__ float4 cf4[NTRI * 5];

    const int lane = threadIdx.x;     // 0..31 (wave32)
    const int tile = blockIdx.x;      // 0..255 -> 16x16 tiles of 16x16 px
    const int tr = tile >> 4;
    const int tc = tile & 15;

    // ---- per-triangle affine coefficients, pre-divided by signed area ----
    for (int k = 0; k < 8; ++k) {
        const int t = lane + (k << 5);
        const float* T = tris + t * 9;
        const float v0x = T[0], v0y = T[1], v0z = T[2];
        const float v1x = T[3], v1y = T[4], v1z = T[5];
        const float v2x = T[6], v2y = T[7], v2z = T[8];
        const float w = (v1x - v0x) * (v2y - v0y) - (v1y - v0y) * (v2x - v0x);
        float4* q = cf4 + t * 5;
        if (w >= 1e-9f) {
            const float inv = 1.0f / w;
            // w1 = pCB/w  (edge a=v2, c=v1)
            const float w1cx = (v1y - v2y) * inv;
            const float w1cy = (v2x - v1x) * inv;
            const float w1c0 = (v2y * (v1x - v2x) - v2x * (v1y - v2y)) * inv;
            // w2 = pCA/w  (edge a=v0, c=v2)
            const float w2cx = (v2y - v0y) * inv;
            const float w2cy = (v0x - v2x) * inv;
            const float w2c0 = (v0y * (v2x - v0x) - v0x * (v2y - v0y)) * inv;
            // w3 = pAB/w  (edge a=v1, c=v0)
            const float w3cx = (v0y - v1y) * inv;
            const float w3cy = (v1x - v0x) * inv;
            const float w3c0 = (v1y * (v0x - v1x) - v1x * (v0y - v1y)) * inv;
            const float* U = uvs + t * 6;
            const float u0 = U[0]*2.f-1.f, s0 = U[1]*2.f-1.f;
            const float u1 = U[2]*2.f-1.f, s1 = U[3]*2.f-1.f;
            const float u2 = U[4]*2.f-1.f, s2 = U[5]*2.f-1.f;
            // z = w1*z0 + w2*z1 + w3*z2
            const float zcx = v0z*w1cx + v1z*w2cx + v2z*w3cx;
            const float zcy = v0z*w1cy + v1z*w2cy + v2z*w3cy;
            const float zc0 = v0z*w1c0 + v1z*w2c0 + v2z*w3c0;
            // texel x = (sum wi*ui + 1) * 0.5 * 1023 ; same for y
            const float ucx = 511.5f*(u0*w1cx + u1*w2cx + u2*w3cx);
            const float ucy = 511.5f*(u0*w1cy + u1*w2cy + u2*w3cy);
            const float uc0 = 511.5f*(u0*w1c0 + u1*w2c0 + u2*w3c0) + 511.5f;
            const float vcx = 511.5f*(s0*w1cx + s1*w2cx + s2*w3cx);
            const float vcy = 511.5f*(s0*w1cy + s1*w2cy + s2*w3cy);
            const float vc0 = 511.5f*(s0*w1c0 + s1*w2c0 + s2*w3c0) + 511.5f;
            q[0] = make_float4(w1cx, w1cy, w1c0, 1.0f);
            q[1] = make_float4(w2cx, w2cy, w2c0, w3cx);
            q[2] = make_float4(w3cy, w3c0, zcx, zcy);
            q[3] = make_float4(zc0, ucx, ucy, uc0);
            q[4] = make_float4(vcx, vcy, vc0, 0.0f);
        } else {
            const float4 z4 = make_float4(0.f, 0.f, 0.f, 0.f);
            q[0] = z4; q[1] = z4; q[2] = z4; q[3] = z4; q[4] = z4;
        }
    }
    __syncthreads();

    // A fragment: A[M,0] = py(M), A[M,1] = 1; K=2,3 slots (hi lanes) = 0.
    // With A's hi-lane slots zeroed, the K=2,3 products vanish regardless of
    // what the (finite) B hi-lane values are, so B needs no masking at all.
    const bool lo = lane < 16;
    const float px = -1.0f + (2.0f/255.0f)*(float)(tc*16 + (lane & 15));
    const float py = 1.0f - (2.0f/255.0f)*(float)(tr*16 + (lane & 15));
    v2f afrag;
    afrag.x = lo ? py   : 0.0f;
    afrag.y = lo ? 1.0f : 0.0f;

    float zb[8], rr[8], gg[8], bbv[8], aa[8];
    #pragma unroll
    for (int i = 0; i < 8; ++i) { zb[i] = -3.0e38f; rr[i]=gg[i]=bbv[i]=aa[i]=0.0f; }

    for (int t = 0; t < NTRI; ++t) {
        const float4* q = cf4 + t * 5;
        const float4 q0 = q[0];
        // uniform skip of degenerate triangles (keeps EXEC all-ones at WMMA)
        if (__builtin_amdgcn_readfirstlane((int)q0.w) == 0) continue;
        const float4 q1 = q[1];
        const float4 q2 = q[2];
        const float4 q3 = q[3];
        const float4 q4 = q[4];

        // B fragment per field: B[0,N] = cy, B[1,N] = c0 + cx*px(N)
        v2f b1, b2, b3, bz, bu, bq;
        b1.x = q0.y;  b1.y = q0.z + q0.x * px;
        b2.x = q1.y;  b2.y = q1.z + q1.x * px;
        b3.x = q2.x;  b3.y = q2.y + q1.w * px;
        bz.x = q2.w;  bz.y = q3.x + q2.z * px;
        bu.x = q3.z;  bu.y = q3.w + q3.y * px;
        bq.x = q4.y;  bq.y = q4.z + q4.x * px;

        v8f z8 = {};
        const v8f d1 = __builtin_amdgcn_wmma_f32_16x16x4_f32(false, afrag, false, b1, (short)0, z8, false, false);
        const v8f d2 = __builtin_amdgcn_wmma_f32_16x16x4_f32(false, afrag, false, b2, (short)0, z8, false, false);
        const v8f d3 = __builtin_amdgcn_wmma_f32_16x16x4_f32(false, afrag, false, b3, (short)0, z8, false, false);
        const v8f dz = __builtin_amdgcn_wmma_f32_16x16x4_f32(false, afrag, false, bz, (short)0, z8, false, false);
        const v8f du = __builtin_amdgcn_wmma_f32_16x16x4_f32(false, afrag, false, bu, (short)0, z8, false, false);
        const v8f dq = __builtin_amdgcn_wmma_f32_16x16x4_f32(false, afrag, false, bq, (short)0, z8, false, false);

        #pragma unroll
        for (int i = 0; i < 8; ++i) {
            const float w1 = d1[i], w2 = d2[i], w3 = d3[i], z = dz[i];
            if (w1 > 0.0f && w2 > 0.0f && w3 > 0.0f && z >= zb[i]) {
                const float x = du[i], y = dq[i];
                const float x0 = floorf(x), y0 = floorf(y);
                const float wx = x - x0, wy = y - y0;
                float R = 0.0f, G = 0.0f, B = 0.0f;
                #pragma unroll
                for (int sub = 0; sub < 4; ++sub) {
                    const float fx = x0 + (float)(sub & 1);
                    const float fy = y0 + (float)(sub >> 1);
                    const bool inb = (fx >= 0.0f) && (fx <= 1023.0f) &&
                                     (fy >= 0.0f) && (fy <= 1023.0f);
                    const unsigned ii = (unsigned)(int)fminf(fmaxf(fx, 0.0f), 1023.0f);
                    const unsigned jj = (unsigned)(int)fminf(fmaxf(fy, 0.0f), 1023.0f);
                    const unsigned idx = (jj << 10) + ii;
                    const float wt = ((sub & 1) ? wx : (1.0f - wx)) *
                                     ((sub >> 1) ? wy : (1.0f - wy));
                    const float mw = (inb ? 1.0f : 0.0f) * wt;
                    R += tex[idx]            * mw;
                    G += tex[1048576u + idx] * mw;
                    B += tex[2097152u + idx] * mw;
                }
                zb[i] = z; rr[i] = R; gg[i] = G; bbv[i] = B; aa[i] = 1.0f;
            }
        }
    }

    // D layout: element i of the accumulator vector is M = i + 8*(lane>=16), N = lane%16
    const unsigned col = (unsigned)(tc * 16 + (lane & 15));
    const unsigned rowBase = (unsigned)(tr * 16 + ((lane >> 4) << 3));
    #pragma unroll
    for (int i = 0; i < 8; ++i) {
        const unsigned p = (rowBase + (unsigned)i) * 256u + col;
        out[p]           = rr[i];
        out[65536u + p]  = gg[i];
        out[131072u + p] = bbv[i];
        out[196608u + p] = aa[i];
    }
}

extern "C" void kernel_launch(void* const* d_in, const int* in_sizes, int n_in,
                              void* d_out, int out_size, void* d_ws, size_t ws_size,
                              hipStream_t stream) {
    (void)in_sizes; (void)n_in; (void)out_size; (void)d_ws; (void)ws_size;
    const float* tris = (const float*)d_in[0];
    const float* uvs  = (const float*)d_in[1];
    const float* tex  = (const float*)d_in[2];
    float* out = (float*)d_out;
    raster16x16_wmma<<<dim3(256), dim3(32), 0, stream>>>(tris, uvs, tex, out);
}